// PointNetSetAbstraction_57037165691605
// MI455X (gfx1250) — compile-verified
//
#include <hip/hip_runtime.h>
#include <hip/hip_bf16.h>
#include <stdint.h>

// ---------------- problem constants (match reference) ----------------
#define BATCH   8
#define NPTS    8192
#define CIN     64
#define NPOINT  2048
#define NSAMPLE 32
#define RAD2    0.04f          // 0.2^2
#define EPSL    1e-5f
#define MROWS   (BATCH * NPOINT * NSAMPLE)   // 524288 rows through the MLP

typedef __attribute__((ext_vector_type(2))) float v2f;
typedef __attribute__((ext_vector_type(8))) float v8f;

// =====================================================================
// 1) Transpose points (B,64,8192) -> (B,8192,64) for contiguous gathers
// =====================================================================
__global__ __launch_bounds__(256) void transpose_points(const float* __restrict__ pin,
                                                        float* __restrict__ pout)
{
    __shared__ float tile[32][33];
    const int b  = blockIdx.z;
    const int n0 = blockIdx.x * 32;
    const int c0 = blockIdx.y * 32;
    const int tx = threadIdx.x, ty = threadIdx.y;      // 32 x 8
    const float* src = pin  + (size_t)b * CIN * NPTS;
    float*       dst = pout + (size_t)b * NPTS * CIN;
    #pragma unroll
    for (int k = 0; k < 32; k += 8)
        tile[ty + k][tx] = src[(size_t)(c0 + ty + k) * NPTS + (n0 + tx)];
    __syncthreads();
    #pragma unroll
    for (int k = 0; k < 32; k += 8)
        dst[(size_t)(n0 + ty + k) * CIN + (c0 + tx)] = tile[tx][ty + k];
}

// =====================================================================
// 2) Farthest point sampling: one block per batch, 8 points per thread,
//    all state in registers; shuffle-tree argmax (wave32).
// =====================================================================
__global__ __launch_bounds__(1024) void fps_kernel(const float* __restrict__ xyz,
                                                   int* __restrict__ fps_idx)
{
    const int b   = blockIdx.x;
    const int tid = threadIdx.x;
    const float* xb = xyz + (size_t)b * 3 * NPTS;
    float px[8], py[8], pz[8], dmin[8];
    #pragma unroll
    for (int k = 0; k < 8; ++k) {
        const int i = tid + k * 1024;          // point i owned by thread i&1023
        px[k] = xb[i];
        py[k] = xb[NPTS + i];
        pz[k] = xb[2 * NPTS + i];
        dmin[k] = 1e10f;
    }
    __shared__ float s_c[3];
    __shared__ float s_wv[32];
    __shared__ int   s_wi[32];
    const int lane = tid & 31;
    const int wv   = tid >> 5;
    int far = 0;
    for (int t = 0; t < NPOINT; ++t) {
        if (tid == 0) fps_idx[b * NPOINT + t] = far;   // record incoming far (matches scan)
        if ((far & 1023) == tid) {                     // owner broadcasts centroid
            const int k = far >> 10;
            s_c[0] = px[k]; s_c[1] = py[k]; s_c[2] = pz[k];
        }
        __syncthreads();
        const float cx = s_c[0], cy = s_c[1], cz = s_c[2];
        float bv = -1.0f; int bi = 0;
        #pragma unroll
        for (int k = 0; k < 8; ++k) {
            const float dx = px[k] - cx, dy = py[k] - cy, dz = pz[k] - cz;
            const float dd = dx * dx + dy * dy + dz * dz;
            dmin[k] = fminf(dmin[k], dd);
            if (dmin[k] > bv) { bv = dmin[k]; bi = tid + k * 1024; }
        }
        #pragma unroll
        for (int off = 16; off > 0; off >>= 1) {       // wave32 argmax, first-index ties
            const float ov = __shfl_down(bv, off, 32);
            const int   oi = __shfl_down(bi, off, 32);
            if (ov > bv || (ov == bv && oi < bi)) { bv = ov; bi = oi; }
        }
        if (lane == 0) { s_wv[wv] = bv; s_wi[wv] = bi; }
        __syncthreads();
        if (wv == 0) {
            float v2 = s_wv[lane]; int i2 = s_wi[lane];
            #pragma unroll
            for (int off = 16; off > 0; off >>= 1) {
                const float ov = __shfl_down(v2, off, 32);
                const int   oi = __shfl_down(i2, off, 32);
                if (ov > v2 || (ov == v2 && oi < i2)) { v2 = ov; i2 = oi; }
            }
            if (lane == 0) s_wi[0] = i2;
        }
        __syncthreads();
        far = s_wi[0];
    }
}

// =====================================================================
// 3) Gather new_xyz (and write the transposed new_xyz output)
// =====================================================================
__global__ void gather_newxyz(const float* __restrict__ xyz, const int* __restrict__ fps,
                              float* __restrict__ nxyz, float* __restrict__ out)
{
    const int i = blockIdx.x * 256 + threadIdx.x;
    if (i >= BATCH * NPOINT) return;
    const int b = i >> 11, p = i & 2047;
    const int j = fps[i];
    const float* xb = xyz + (size_t)b * 3 * NPTS;
    const float x = xb[j], y = xb[NPTS + j], z = xb[2 * NPTS + j];
    nxyz[(size_t)i * 3 + 0] = x;
    nxyz[(size_t)i * 3 + 1] = y;
    nxyz[(size_t)i * 3 + 2] = z;
    out[((size_t)b * 3 + 0) * NPOINT + p] = x;   // (B,3,NPOINT) output layout
    out[((size_t)b * 3 + 1) * NPOINT + p] = y;
    out[((size_t)b * 3 + 2) * NPOINT + p] = z;
}

// =====================================================================
// 4) Ball query: first NSAMPLE indices (ascending) within RAD2; pad with
//    the first hit. LDS-tiled scan with early exit.
// =====================================================================
__global__ __launch_bounds__(128) void query_ball(const float* __restrict__ xyz,
                                                  const float* __restrict__ nxyz,
                                                  int* __restrict__ ball)
{
    __shared__ float tx[1024], ty[1024], tz[1024];
    const int b = blockIdx.y;
    const int q = blockIdx.x * 128 + threadIdx.x;
    const size_t bq = (size_t)b * NPOINT + q;
    const float qx = nxyz[bq * 3 + 0], qy = nxyz[bq * 3 + 1], qz = nxyz[bq * 3 + 2];
    int* out = ball + bq * NSAMPLE;
    const float* xb = xyz + (size_t)b * 3 * NPTS;
    int cnt = 0;
    for (int tile = 0; tile < NPTS / 1024; ++tile) {
        __syncthreads();
        for (int j = threadIdx.x; j < 1024; j += 128) {
            const int i = tile * 1024 + j;
            tx[j] = xb[i]; ty[j] = xb[NPTS + i]; tz[j] = xb[2 * NPTS + i];
        }
        __syncthreads();
        if (cnt < NSAMPLE) {
            for (int j = 0; j < 1024; ++j) {
                const float dx = tx[j] - qx, dy = ty[j] - qy, dz = tz[j] - qz;
                const float dd = dx * dx + dy * dy + dz * dz;
                if (dd <= RAD2) {
                    out[cnt++] = tile * 1024 + j;
                    if (cnt == NSAMPLE) break;
                }
            }
        }
    }
    const int f = (cnt > 0) ? out[0] : 0;   // query point itself always hits (d=0)
    for (; cnt < NSAMPLE; ++cnt) out[cnt] = f;
}

// =====================================================================
// 5) Build feat0 rows: [xyz_j - new_xyz_p (3) | points_j (64) | pad(1)]
//    K padded to 68 so the WMMA K-loop runs in chunks of 4 with NO guards.
// =====================================================================
__global__ __launch_bounds__(128) void gather_feat(const float* __restrict__ xyz,
                                                   const float* __restrict__ nxyz,
                                                   const float* __restrict__ pT,
                                                   const int* __restrict__ ball,
                                                   float* __restrict__ bufA)
{
    const size_t r  = (size_t)blockIdx.x * 128 + threadIdx.x;   // 0..MROWS-1
    const size_t bp = r >> 5;
    const int    b  = (int)(bp >> 11);
    const int    j  = ball[r];
    float* o = bufA + r * 68;
    const float* xb = xyz + (size_t)b * 3 * NPTS;
    o[0] = xb[j]            - nxyz[bp * 3 + 0];
    o[1] = xb[NPTS + j]     - nxyz[bp * 3 + 1];
    o[2] = xb[2 * NPTS + j] - nxyz[bp * 3 + 2];
    const float* src = pT + ((size_t)b * NPTS + j) * CIN;
    #pragma unroll 8
    for (int c = 0; c < CIN; ++c) o[3 + c] = src[c];
    o[67] = 0.0f;
}

// =====================================================================
// 5b) Zero-pad W1 (64x67) into Wp1 (64x68, col 67 = 0)
// =====================================================================
__global__ void pack_w1(const float* __restrict__ W, float* __restrict__ Wp)
{
    const int i = blockIdx.x * 256 + threadIdx.x;      // 64*68 = 4352
    if (i >= 64 * 68) return;
    const int n = i / 68, k = i % 68;
    Wp[i] = (k < 67) ? W[n * 67 + k] : 0.0f;
}

// =====================================================================
// 6) WMMA GEMM: Y[M,N] = relu_bn?(X)[M,Kp] * Wp[N,Kp]^T + bias
//    Kp = K4*4 (zero padded), compile-time -> no guards, full unroll.
//    Weights staged in LDS via CDNA5 async-to-LDS DMA; B operands come
//    from ds_load_b64. Wave = 16-row strip x NT 16x16 f32 WMMA tiles.
//    A layout: lane M = lane&15, K-pair = (lane>>4)*2, VGPR -> K in pair.
//    B layout: lane N = lane&15, VGPR j -> K = j + (lane<16?0:2).
//    C layout: lane N = lane&15, VGPR v -> M = v + (lane<16?0:8).
// =====================================================================
template <int K4, int NT, bool NORM>
__global__ __launch_bounds__(256) void gemm_kernel(const float* __restrict__ X, int sx,
                                                   const float* __restrict__ Wp,
                                                   const float* __restrict__ bias,
                                                   const float* __restrict__ scale,
                                                   const float* __restrict__ shift,
                                                   float* __restrict__ Y, int sy)
{
    constexpr int K   = K4 * 4;
    constexpr int NCH = NT * 16;

    __shared__ float sW[NCH * K];          // <= 128*64*4 = 32 KB
    __shared__ float sSc[K];
    __shared__ float sSh[K];

    // ---- stage weights into LDS with async DMA (GLOBAL_LOAD_ASYNC_TO_LDS_B32) ----
    #pragma unroll
    for (int i = threadIdx.x; i < NCH * K; i += 256) {
        const unsigned           ldsOff = (unsigned)(uintptr_t)(&sW[i]);
        const unsigned long long gaddr  = (unsigned long long)(uintptr_t)(Wp + i);
        asm volatile("global_load_async_to_lds_b32 %0, %1, off"
                     :: "v"(ldsOff), "v"(gaddr) : "memory");
    }
    if constexpr (NORM) {
        for (int i = threadIdx.x; i < K; i += 256) { sSc[i] = scale[i]; sSh[i] = shift[i]; }
    }
    asm volatile("s_wait_asynccnt 0x0" ::: "memory");
    __syncthreads();

    const int    lane    = threadIdx.x & 31;
    const int    wave    = threadIdx.x >> 5;                 // 8 waves -> 128 rows/block
    const size_t rowBase = (size_t)blockIdx.x * 128 + (size_t)wave * 16;
    const int    mrow    = lane & 15;
    const int    khalf   = (lane >> 4) * 2;                  // 0 or 2

    v8f acc[NT];
    const v8f zero = {0.f, 0.f, 0.f, 0.f, 0.f, 0.f, 0.f, 0.f};
    #pragma unroll
    for (int t = 0; t < NT; ++t) acc[t] = zero;

    const float* xrow = X + (rowBase + mrow) * (size_t)sx;
    __builtin_prefetch(xrow, 0, 3);                          // global_prefetch_b8

    #pragma unroll
    for (int kc = 0; kc < K4; ++kc) {
        const int kk = kc * 4 + khalf;                       // even -> 8B aligned
        v2f a = *(const v2f*)(xrow + kk);
        if constexpr (NORM) {
            a.x = fmaxf(fmaf(a.x, sSc[kk],     sSh[kk]),     0.f);
            a.y = fmaxf(fmaf(a.y, sSc[kk + 1], sSh[kk + 1]), 0.f);
        }
        #pragma unroll
        for (int t = 0; t < NT; ++t) {
            const int n = t * 16 + mrow;
            const v2f bb = *(const v2f*)(&sW[n * K + kk]);   // ds_load_b64
            acc[t] = __builtin_amdgcn_wmma_f32_16x16x4_f32(
                /*neg_a=*/false, a, /*neg_b=*/false, bb,
                /*c_mod=*/(short)0, acc[t], /*reuse_a=*/false, /*reuse_b=*/false);
        }
    }

    const int mOff = (lane >> 4) << 3;                       // 0 or 8
    #pragma unroll
    for (int t = 0; t < NT; ++t) {
        const int n = t * 16 + mrow;
        const float bs = bias[n];
        float* yb = Y + (rowBase + mOff) * (size_t)sy + n;
        #pragma unroll
        for (int v = 0; v < 8; ++v) yb[(size_t)v * sy] = acc[t][v] + bs;
    }
}

// =====================================================================
// 7) Per-channel sum / sumsq (for batch-norm over all M rows)
// =====================================================================
template <int C>
__global__ __launch_bounds__(256) void reduce_stats(const float* __restrict__ Y, int sy,
                                                    float* __restrict__ sums,
                                                    float* __restrict__ sumsq)
{
    const int tid = threadIdx.x;
    const int c   = tid % C;
    const int g   = tid / C;
    const int G   = 256 / C;
    float s = 0.f, s2 = 0.f;
    const size_t r1 = (size_t)(blockIdx.x + 1) * 2048;       // MROWS/2048 = 256 blocks
    for (size_t r = (size_t)blockIdx.x * 2048 + g; r < r1; r += G) {
        const float v = Y[r * (size_t)sy + c];
        s += v; s2 += v * v;
    }
    __shared__ float ls[256], lq[256];
    ls[tid] = s; lq[tid] = s2;
    __syncthreads();
    if (tid < C) {
        for (int gg = 1; gg < G; ++gg) { s += ls[gg * C + tid]; s2 += lq[gg * C + tid]; }
        atomicAdd(&sums[tid], s);
        atomicAdd(&sumsq[tid], s2);
    }
}

__global__ void finalize_stats(const float* __restrict__ sums, const float* __restrict__ sumsq,
                               const float* __restrict__ gamma, const float* __restrict__ beta,
                               int C, float* __restrict__ scale, float* __restrict__ shift)
{
    const int c = threadIdx.x;
    if (c < C) {
        const float invM = 1.0f / (float)MROWS;
        const float mean = sums[c] * invM;
        const float var  = sumsq[c] * invM - mean * mean;
        const float sc   = gamma[c] * rsqrtf(var + EPSL);
        scale[c] = sc;
        shift[c] = beta[c] - mean * sc;
    }
}

// =====================================================================
// 8) Final BN+ReLU+max over the 32 samples, transposed store (B,128,NPOINT)
// =====================================================================
__global__ __launch_bounds__(128) void max_kernel(const float* __restrict__ Y,
                                                  const float* __restrict__ scale,
                                                  const float* __restrict__ shift,
                                                  float* __restrict__ out)
{
    const int    c  = threadIdx.x;           // 128 channels
    const size_t bp = blockIdx.x;            // b*NPOINT + p
    const float  sc = scale[c], sh = shift[c];
    const float* y  = Y + bp * NSAMPLE * 128 + c;
    float m = -3.0e38f;
    #pragma unroll
    for (int s = 0; s < NSAMPLE; ++s)
        m = fmaxf(m, fmaf(y[(size_t)s * 128], sc, sh));
    m = fmaxf(m, 0.0f);
    const int b = (int)(bp >> 11);
    const int p = (int)(bp & 2047);
    out[((size_t)b * 128 + c) * NPOINT + p] = m;
}

// =====================================================================
// Host-side orchestration
// =====================================================================
extern "C" void kernel_launch(void* const* d_in, const int* in_sizes, int n_in,
                              void* d_out, int out_size, void* d_ws, size_t ws_size,
                              hipStream_t stream)
{
    (void)in_sizes; (void)n_in; (void)out_size; (void)ws_size;
    const float* xyz    = (const float*)d_in[0];
    const float* points = (const float*)d_in[1];
    const float* W1 = (const float*)d_in[2];
    const float* b1 = (const float*)d_in[3];
    const float* g1 = (const float*)d_in[4];
    const float* be1= (const float*)d_in[5];
    const float* W2 = (const float*)d_in[6];
    const float* b2 = (const float*)d_in[7];
    const float* g2 = (const float*)d_in[8];
    const float* be2= (const float*)d_in[9];
    const float* W3 = (const float*)d_in[10];
    const float* b3 = (const float*)d_in[11];
    const float* g3 = (const float*)d_in[12];
    const float* be3= (const float*)d_in[13];
    float* out = (float*)d_out;

    // ---- workspace carve-up (all offsets 256B-aligned) ----
    char* ws = (char*)d_ws;
    int*   fps   = (int*)  (ws + 0);                 //  65536 B : B*NPOINT int
    int*   ball  = (int*)  (ws + 65536);             //   2 MiB : B*NPOINT*NSAMPLE int
    float* nxyz  = (float*)(ws + 2162688);           // 192 KiB : B*NPOINT*3 f32
    float* stats = (float*)(ws + 2359296);           //   6 KiB : 12 arrays x 128 f32
    float* Wp1   = (float*)(ws + 2365440);           //  17 KiB: 64x68 padded W1
    float* pT    = (float*)(ws + 2382848);           //  16 MiB : (B,NPTS,CIN)
    float* bufA  = (float*)(ws + 19160064);          // 142.6 MB: MROWS x 68
    float* bufB  = (float*)(ws + 161766400);         // 268.4 MB: MROWS x 128

    hipMemsetAsync(stats, 0, 12 * 128 * sizeof(float), stream);

    transpose_points<<<dim3(NPTS / 32, CIN / 32, BATCH), dim3(32, 8), 0, stream>>>(points, pT);
    pack_w1<<<(64 * 68 + 255) / 256, 256, 0, stream>>>(W1, Wp1);
    fps_kernel<<<BATCH, 1024, 0, stream>>>(xyz, fps);
    gather_newxyz<<<(BATCH * NPOINT + 255) / 256, 256, 0, stream>>>(xyz, fps, nxyz, out);
    query_ball<<<dim3(NPOINT / 128, BATCH), 128, 0, stream>>>(xyz, nxyz, ball);
    gather_feat<<<MROWS / 128, 128, 0, stream>>>(xyz, nxyz, pT, ball, bufA);

    // Layer 1: feat0(67->68 padded) -> bufB(64, stride 128)
    gemm_kernel<17, 4, false><<<MROWS / 128, 256, 0, stream>>>(bufA, 68, Wp1, b1,
                                                               nullptr, nullptr, bufB, 128);
    reduce_stats<64><<<256, 256, 0, stream>>>(bufB, 128, stats + 0, stats + 128);
    finalize_stats<<<1, 64, 0, stream>>>(stats + 0, stats + 128, g1, be1, 64,
                                         stats + 256, stats + 384);
    // Layer 2: bn_relu(bufB)(64) -> bufA(64, stride 68)
    gemm_kernel<16, 4, true><<<MROWS / 128, 256, 0, stream>>>(bufB, 128, W2, b2,
                                                              stats + 256, stats + 384, bufA, 68);
    reduce_stats<64><<<256, 256, 0, stream>>>(bufA, 68, stats + 512, stats + 640);
    finalize_stats<<<1, 64, 0, stream>>>(stats + 512, stats + 640, g2, be2, 64,
                                         stats + 768, stats + 896);
    // Layer 3: bn_relu(bufA)(64) -> bufB(128, stride 128)
    gemm_kernel<16, 8, true><<<MROWS / 128, 256, 0, stream>>>(bufA, 68, W3, b3,
                                                              stats + 768, stats + 896, bufB, 128);
    reduce_stats<128><<<256, 256, 0, stream>>>(bufB, 128, stats + 1024, stats + 1152);
    finalize_stats<<<1, 128, 0, stream>>>(stats + 1024, stats + 1152, g3, be3, 128,
                                          stats + 1280, stats + 1408);
    // BN + ReLU + max over samples, transposed store after the new_xyz block
    max_kernel<<<BATCH * NPOINT, 128, 0, stream>>>(bufB, stats + 1280, stats + 1408,
                                                   out + (size_t)BATCH * 3 * NPOINT);
}